// LightGCN_20873541059099
// MI455X (gfx1250) — compile-verified
//
#include <hip/hip_runtime.h>
#include <hip/hip_bf16.h>

// ---------------------------------------------------------------------------
// LightGCN propagation for MI455X (gfx1250, wave32).
//
//  ego  = concat(user_emb, item_emb)            [N=150000, 64] f32
//  3x:  nxt[r] += vals[e] * ego[cols[e]]  (COO scatter SpMM, 2.4M edges)
//       acc += nxt ; ego = nxt
//  out  = acc / 3
//
// Memory-bound: embedding table (38.4MB) fits in 192MB L2, so gather/scatter
// lives in L2; HBM sees only the 28.8MB/layer edge stream + dense sweeps.
// Scatter uses native global_atomic_add_f32 (unsafeAtomicAdd), 16 lanes /
// edge, float4 per lane (coalesced 256B rows). Epilogue (x 1/3) runs on the
// matrix pipe via V_WMMA_F32_16X16X4_F32 with a scaled shifted-identity B.
// ---------------------------------------------------------------------------

typedef float v2f __attribute__((ext_vector_type(2)));
typedef float v8f __attribute__((ext_vector_type(8)));

#define N_NODES 150000
#define USER_NUM 100000
#define EMB 64

// ---- init: ego = concat(user,item); acc = 0; first scatter dest = 0 --------
__global__ void lgcn_init(const float* __restrict__ user,
                          const float* __restrict__ item,
                          float* __restrict__ ego,
                          float* __restrict__ acc,
                          float* __restrict__ dst0,
                          int total, int u64) {
  const int stride = gridDim.x * blockDim.x;
  for (int i = blockIdx.x * blockDim.x + threadIdx.x; i < total; i += stride) {
    ego[i]  = (i < u64) ? user[i] : item[i - u64];
    acc[i]  = 0.0f;
    dst0[i] = 0.0f;
  }
}

// ---- SpMM scatter: 16 lanes per edge, float4 per lane ----------------------
__global__ void lgcn_scatter(const int*   __restrict__ rows,
                             const int*   __restrict__ cols,
                             const float* __restrict__ vals,
                             const float* __restrict__ ego,
                             float*       __restrict__ nxt,
                             int n_edges) {
  const int stride  = gridDim.x * blockDim.x;   // multiple of 16
  const int estride = stride >> 4;              // edge stride
  const int total   = n_edges << 4;
  for (int t = blockIdx.x * blockDim.x + threadIdx.x; t < total; t += stride) {
    const int e = t >> 4;
    const int q = t & 15;                        // dim quad 0..15
    if (q == 0) {                                // gfx1250 global_prefetch_b8
      __builtin_prefetch(cols + e + estride, 0, 0);
      __builtin_prefetch(rows + e + estride, 0, 0);
      __builtin_prefetch(vals + e + estride, 0, 0);
    }
    const int   col = cols[e];
    const int   row = rows[e];
    const float v   = vals[e];
    const float4 g  = *reinterpret_cast<const float4*>(ego + (size_t)col * EMB + q * 4);
    float* o = nxt + (size_t)row * EMB + q * 4;
    unsafeAtomicAdd(o + 0, v * g.x);             // global_atomic_add_f32
    unsafeAtomicAdd(o + 1, v * g.y);
    unsafeAtomicAdd(o + 2, v * g.z);
    unsafeAtomicAdd(o + 3, v * g.w);
  }
}

// ---- acc += dst ; optionally zero the buffer reused as next layer's dst ----
__global__ void lgcn_add_clear(float* __restrict__ acc,
                               const float* __restrict__ dst,
                               float* __restrict__ clearbuf,
                               int total) {
  const int stride = gridDim.x * blockDim.x;
  for (int i = blockIdx.x * blockDim.x + threadIdx.x; i < total; i += stride) {
    acc[i] += dst[i];
    if (clearbuf) clearbuf[i] = 0.0f;
  }
}

// ---- epilogue: acc *= 1/3 in place via V_WMMA_F32_16X16X4_F32 --------------
// Each wave owns a 16-row x 16-col tile:  D = sum_j A_j (16x4) * B_j (4x16),
// A_j = acc[:, c0+4j : c0+4j+4],  B_j[k][n] = (n == 4j+k) ? 1/3 : 0.
// f32 A layout: v0 = K{0|2}, v1 = K{1|3} (lanes 0-15 | 16-31), M = lane%16.
// B mirrors A (rows striped across lanes); C/D: VGPR r -> M = r + 8*half.
__global__ void lgcn_mean_wmma(float* __restrict__ acc) {
  const int lane = threadIdx.x & 31;
  const int wave = threadIdx.x >> 5;          // 0..3 -> 16-col slice
  const int c0   = wave * 16;
  const int half = lane >> 4;                 // 0: lanes 0-15, 1: lanes 16-31
  const int n    = lane & 15;                 // col for B/D, row for A
  float* base    = acc + (size_t)blockIdx.x * 16 * EMB;

  const float s = 1.0f / 3.0f;
  v8f d = {0.f, 0.f, 0.f, 0.f, 0.f, 0.f, 0.f, 0.f};
#pragma unroll
  for (int j = 0; j < 4; ++j) {
    const int k0 = half ? 2 : 0;
    const int k1 = half ? 3 : 1;
    v2f a, b;
    a.x = base[n * EMB + c0 + 4 * j + k0];
    a.y = base[n * EMB + c0 + 4 * j + k1];
    b.x = (n == 4 * j + k0) ? s : 0.0f;
    b.y = (n == 4 * j + k1) ? s : 0.0f;
    d = __builtin_amdgcn_wmma_f32_16x16x4_f32(false, a, false, b,
                                              (short)0, d, false, false);
  }
#pragma unroll
  for (int r = 0; r < 8; ++r)
    base[(r + half * 8) * EMB + c0 + n] = d[r];
}

// ---------------------------------------------------------------------------
extern "C" void kernel_launch(void* const* d_in, const int* in_sizes, int n_in,
                              void* d_out, int out_size, void* d_ws, size_t ws_size,
                              hipStream_t stream) {
  const float* user = (const float*)d_in[0];
  const float* item = (const float*)d_in[1];
  const int*   rows = (const int*)d_in[2];
  const int*   cols = (const int*)d_in[3];
  const float* vals = (const float*)d_in[4];
  const int n_edges = in_sizes[2];

  const int total = N_NODES * EMB;             // 9.6M floats
  float* ws0 = (float*)d_ws;                   // ping
  float* ws1 = ws0 + (size_t)total;            // pong   (needs 76.8MB of ws)
  float* acc = (float*)d_out;                  // layer sum accumulator

  // ego = concat ; acc = 0 ; layer-0 destination (ws1) = 0
  lgcn_init<<<2048, 256, 0, stream>>>(user, item, ws0, acc, ws1, total,
                                      USER_NUM * EMB);

  for (int l = 0; l < 3; ++l) {
    float* src = (l & 1) ? ws1 : ws0;          // l=0:ws0  l=1:ws1  l=2:ws0
    float* dst = (l & 1) ? ws0 : ws1;
    lgcn_scatter<<<3072, 256, 0, stream>>>(rows, cols, vals, src, dst, n_edges);
    // acc += dst ; zero src (it becomes next layer's destination)
    float* clearbuf = (l < 2) ? src : nullptr;
    lgcn_add_clear<<<2048, 256, 0, stream>>>(acc, dst, clearbuf, total);
  }

  // out = acc / 3, in place on d_out, via the WMMA pipe (150000 % 16 == 0)
  lgcn_mean_wmma<<<N_NODES / 16, 128, 0, stream>>>(acc);
}